// StaticGCN_85942295593306
// MI455X (gfx1250) — compile-verified
//
#include <hip/hip_runtime.h>
#include <math.h>

#define N_NODES 100000
#define F_IN    256
#define HID1    128
#define HID2    64
#define LN_EPS  1e-5f

typedef __attribute__((ext_vector_type(2))) float v2f;
typedef __attribute__((ext_vector_type(8))) float v8f;

// Hardware FP32 atomic add (no return -> STOREcnt-tracked, no CAS loop).
// s_endpgm performs an implicit wait-idle, so pending atomics drain at wave end.
static __device__ __forceinline__ void atomic_add_f32(float* p, float v) {
    asm volatile("global_atomic_add_f32 %0, %1, off" :: "v"(p), "v"(v) : "memory");
}

// ---------------- degree / normalization ----------------
__global__ void k_init_deg(float* __restrict__ deg, int n) {
    int i = blockIdx.x * blockDim.x + threadIdx.x;
    if (i < n) deg[i] = 1.0f;              // self-loop contributes 1
}

__global__ void k_deg_accum(const int* __restrict__ dst, float* __restrict__ deg, int e) {
    int i = blockIdx.x * blockDim.x + threadIdx.x;
    if (i < e) atomic_add_f32(&deg[dst[i]], 1.0f);
}

__global__ void k_rsqrt_inplace(float* __restrict__ d, int n) {
    int i = blockIdx.x * blockDim.x + threadIdx.x;
    if (i < n) d[i] = rsqrtf(d[i]);
}

// ---------------- fp32 WMMA GEMM: Y[N,OUTC] = X[N,K] @ W[K,OUTC] ----------------
// grid.x = N/16 row tiles, blockDim.x = (OUTC/16)*32 (one wave per 16-col tile).
template<int K, int OUTC>
__global__ void k_gemm_wmma_f32(const float* __restrict__ X,
                                const float* __restrict__ W,
                                float* __restrict__ Y) {
    __shared__ float lx[16][K + 1];        // +1 pad: stride%64==1 -> conflict-free
    const int nthreads = (OUTC / 16) * 32;
    const int t    = threadIdx.x;
    const int row0 = blockIdx.x * 16;

    // Stage 16xK tile of X into LDS, fully coalesced.
    for (int idx = t; idx < 16 * K; idx += nthreads) {
        int r = idx / K, c = idx % K;
        lx[r][c] = X[(size_t)(row0 + r) * K + c];
    }
    __syncthreads();

    const int wave = t >> 5;
    const int lane = t & 31;
    const int lm   = lane & 15;            // M (A) / N (B,D) index
    const int kh   = (lane >> 4) << 1;     // lanes 0-15: K=0,1 ; lanes 16-31: K=2,3
    const int col0 = wave * 16;

    v8f acc = {};
#pragma unroll 4
    for (int k = 0; k < K; k += 4) {
        v2f a, b;
        a.x = lx[lm][k + kh];
        a.y = lx[lm][k + kh + 1];
        b.x = W[(size_t)(k + kh)     * OUTC + col0 + lm];
        b.y = W[(size_t)(k + kh + 1) * OUTC + col0 + lm];
        acc = __builtin_amdgcn_wmma_f32_16x16x4_f32(
                  false, a, false, b, (short)0, acc, false, false);
    }

    // D layout: VGPR j -> row j (lanes 0-15) / row j+8 (lanes 16-31), col = lane%16
    const int rbase = (lane >> 4) * 8;
#pragma unroll
    for (int i = 0; i < 8; i++)
        Y[(size_t)(row0 + rbase + i) * OUTC + col0 + lm] = acc[i];
}

// ---------------- aggregation init: agg = dinv^2 * h + bias ----------------
template<int C>
__global__ void k_agg_init(const float* __restrict__ h, const float* __restrict__ dinv,
                           const float* __restrict__ bias, float* __restrict__ agg, int n) {
    int i = blockIdx.x * blockDim.x + threadIdx.x;
    if (i >= n * C) return;
    int node = i / C, c = i % C;
    float di = dinv[node];
    agg[i] = h[i] * di * di + bias[c];
}

// ---------------- per-edge scatter-add: one wave32 per edge ----------------
template<int C>   // C = 128 -> float4/lane ; C = 64 -> float2/lane
__global__ void k_edge_agg(const int* __restrict__ ei, const float* __restrict__ h,
                           const float* __restrict__ dinv, float* __restrict__ agg, int e) {
    int lane = threadIdx.x & 31;
    int edge = blockIdx.x * (blockDim.x >> 5) + (threadIdx.x >> 5);
    if (edge >= e) return;
    int s = ei[edge];
    int d = ei[e + edge];
    float w = dinv[s] * dinv[d];
    constexpr int V = C / 32;
    const float* hp = h   + (size_t)s * C + lane * V;
    float*       ap = agg + (size_t)d * C + lane * V;
    if constexpr (V == 4) {
        float4 hv = *reinterpret_cast<const float4*>(hp);
        atomic_add_f32(ap + 0, w * hv.x);
        atomic_add_f32(ap + 1, w * hv.y);
        atomic_add_f32(ap + 2, w * hv.z);
        atomic_add_f32(ap + 3, w * hv.w);
    } else {
        float2 hv = *reinterpret_cast<const float2*>(hp);
        atomic_add_f32(ap + 0, w * hv.x);
        atomic_add_f32(ap + 1, w * hv.y);
    }
}

// ---------------- LayerNorm + ReLU over C=128 (wave per node) ----------------
__global__ void k_ln_relu_128(const float* __restrict__ in, const float* __restrict__ g,
                              const float* __restrict__ be, float* __restrict__ out, int n) {
    int lane = threadIdx.x & 31;
    int node = blockIdx.x * (blockDim.x >> 5) + (threadIdx.x >> 5);
    if (node >= n) return;
    const float4 v = *reinterpret_cast<const float4*>(in + (size_t)node * 128 + lane * 4);
    float s  = v.x + v.y + v.z + v.w;
    float ss = v.x * v.x + v.y * v.y + v.z * v.z + v.w * v.w;
#pragma unroll
    for (int m = 16; m > 0; m >>= 1) {
        s  += __shfl_xor(s,  m, 32);
        ss += __shfl_xor(ss, m, 32);
    }
    float mu  = s * (1.0f / 128.0f);
    float var = ss * (1.0f / 128.0f) - mu * mu;
    float r   = rsqrtf(var + LN_EPS);
    float4 gv = *reinterpret_cast<const float4*>(g  + lane * 4);
    float4 bv = *reinterpret_cast<const float4*>(be + lane * 4);
    float4 o;
    o.x = fmaxf((v.x - mu) * r * gv.x + bv.x, 0.0f);
    o.y = fmaxf((v.y - mu) * r * gv.y + bv.y, 0.0f);
    o.z = fmaxf((v.z - mu) * r * gv.z + bv.z, 0.0f);
    o.w = fmaxf((v.w - mu) * r * gv.w + bv.w, 0.0f);
    *reinterpret_cast<float4*>(out + (size_t)node * 128 + lane * 4) = o;
}

// ------- LayerNorm + ReLU + readout (C=64): out = sigmoid(relu(LN(h))@Wr + br) -------
__global__ void k_ln_relu_readout_64(const float* __restrict__ in, const float* __restrict__ g,
                                     const float* __restrict__ be, const float* __restrict__ wr,
                                     const float* __restrict__ br, float* __restrict__ out, int n) {
    int lane = threadIdx.x & 31;
    int node = blockIdx.x * (blockDim.x >> 5) + (threadIdx.x >> 5);
    if (node >= n) return;
    const float2 v = *reinterpret_cast<const float2*>(in + (size_t)node * 64 + lane * 2);
    float s  = v.x + v.y;
    float ss = v.x * v.x + v.y * v.y;
#pragma unroll
    for (int m = 16; m > 0; m >>= 1) {
        s  += __shfl_xor(s,  m, 32);
        ss += __shfl_xor(ss, m, 32);
    }
    float mu  = s * (1.0f / 64.0f);
    float var = ss * (1.0f / 64.0f) - mu * mu;
    float r   = rsqrtf(var + LN_EPS);
    float2 gv = *reinterpret_cast<const float2*>(g  + lane * 2);
    float2 bv = *reinterpret_cast<const float2*>(be + lane * 2);
    float2 wv = *reinterpret_cast<const float2*>(wr + lane * 2);
    float y0 = fmaxf((v.x - mu) * r * gv.x + bv.x, 0.0f);
    float y1 = fmaxf((v.y - mu) * r * gv.y + bv.y, 0.0f);
    float dot = y0 * wv.x + y1 * wv.y;
#pragma unroll
    for (int m = 16; m > 0; m >>= 1) dot += __shfl_xor(dot, m, 32);
    if (lane == 0) {
        float z = dot + br[0];
        out[node] = 1.0f / (1.0f + expf(-z));
    }
}

// ---------------- launch ----------------
extern "C" void kernel_launch(void* const* d_in, const int* in_sizes, int n_in,
                              void* d_out, int out_size, void* d_ws, size_t ws_size,
                              hipStream_t stream) {
    const float* x   = (const float*)d_in[0];
    const int*   ei  = (const int*)d_in[1];      // [2, E]
    const float* W1  = (const float*)d_in[2];
    const float* b1  = (const float*)d_in[3];
    const float* g1  = (const float*)d_in[4];
    const float* be1 = (const float*)d_in[5];
    const float* W2  = (const float*)d_in[6];
    const float* b2  = (const float*)d_in[7];
    const float* g2  = (const float*)d_in[8];
    const float* be2 = (const float*)d_in[9];
    const float* Wr  = (const float*)d_in[10];
    const float* br  = (const float*)d_in[11];
    float* out = (float*)d_out;

    const int n = N_NODES;
    const int e = in_sizes[1] / 2;

    // Workspace layout (with reuse): deg[N] | buf1[N*128] | buf2[N*128]
    float* deg  = (float*)d_ws;
    float* buf1 = deg + n;
    float* buf2 = buf1 + (size_t)n * HID1;
    float* h1   = buf1;                         // GEMM1 out
    float* agg1 = buf2;                         // layer-1 aggregation
    float* x2   = buf1;                         // LN1 out (h1 dead)
    float* h2   = buf2;                         // GEMM2 out (agg1 dead), N*64
    float* agg2 = buf2 + (size_t)n * HID2;      // layer-2 aggregation, N*64

    // degree -> dinv
    k_init_deg  <<<(n + 255) / 256, 256, 0, stream>>>(deg, n);
    k_deg_accum <<<(e + 255) / 256, 256, 0, stream>>>(ei + e, deg, e);
    k_rsqrt_inplace<<<(n + 255) / 256, 256, 0, stream>>>(deg, n);

    // layer 1
    k_gemm_wmma_f32<F_IN, HID1><<<n / 16, (HID1 / 16) * 32, 0, stream>>>(x, W1, h1);
    k_agg_init<HID1><<<((size_t)n * HID1 + 255) / 256, 256, 0, stream>>>(h1, deg, b1, agg1, n);
    k_edge_agg<HID1><<<(e + 7) / 8, 256, 0, stream>>>(ei, h1, deg, agg1, e);
    k_ln_relu_128<<<(n + 7) / 8, 256, 0, stream>>>(agg1, g1, be1, x2, n);

    // layer 2
    k_gemm_wmma_f32<HID1, HID2><<<n / 16, (HID2 / 16) * 32, 0, stream>>>(x2, W2, h2);
    k_agg_init<HID2><<<((size_t)n * HID2 + 255) / 256, 256, 0, stream>>>(h2, deg, b2, agg2, n);
    k_edge_agg<HID2><<<(e + 7) / 8, 256, 0, stream>>>(ei, h2, deg, agg2, e);

    // readout
    k_ln_relu_readout_64<<<(n + 7) / 8, 256, 0, stream>>>(agg2, g2, be2, Wr, br, out, n);
}